// Skeleton_37443524887229
// MI455X (gfx1250) — compile-verified
//
#include <hip/hip_runtime.h>
#include <hip/hip_bf16.h>

// ---------------------------------------------------------------------------
// SMPL forward kinematics, MI455X (gfx1250).
// Memory-bound (~77MB traffic, ~0.4 GFLOP) -> optimize the data path:
//   GLOBAL_LOAD_ASYNC_TO_LDS_B128 / GLOBAL_STORE_ASYNC_FROM_LDS_B128 (ASYNCcnt)
//   stage 128-element slabs with perfectly coalesced 16B/lane transfers.
// fk_kernel_full: branch-free hot path (all blocks for B % 128 == 0).
// fk_kernel_tail: guarded remainder path, launched only if needed (never for
//                 the fixed B = 131072 harness).
// ---------------------------------------------------------------------------

typedef int v4i __attribute__((vector_size(16)));   // matches builtin's V4i

#define AS1 __attribute__((address_space(1)))
#define AS3 __attribute__((address_space(3)))

#if defined(__has_builtin)
#if __has_builtin(__builtin_amdgcn_global_load_async_to_lds_b128) && \
    __has_builtin(__builtin_amdgcn_global_store_async_from_lds_b128)
#define USE_ASYNC_LDS 1
#endif
#endif
#ifndef USE_ASYNC_LDS
#define USE_ASYNC_LDS 0
#endif

static __device__ __forceinline__ void wait_async0() {
#if defined(__has_builtin) && __has_builtin(__builtin_amdgcn_s_wait_asynccnt)
    __builtin_amdgcn_s_wait_asynccnt(0);
#else
    asm volatile("s_wait_asynccnt 0" ::: "memory");
#endif
}

// Static SMPL kinematic tree (parents[i] < i) — static in the reference too.
constexpr int kJ = 24;
constexpr int kPar[kJ] = {-1, 0, 0, 0, 1, 2, 3, 4, 5, 6, 7, 8, 9, 9, 9,
                          12, 13, 14, 16, 17, 18, 19, 20, 21};

#define EPB   128   // batch elements per block (= threads per block, 4 waves)
#define ROWF  72    // floats per element (24 joints * 3)
#define ROWP  76    // padded LDS row stride in dwords (16B-aligned rows)
#define CPT   18    // float4 chunks per thread: (EPB*ROWF/4)/EPB

// one float4 chunk: global dword offset c*4 <-> lds[(c/18)*ROWP + (c%18)*4]
static __device__ __forceinline__ void stage_in(const float* gsrc, float* lds, int c) {
    const int e = c / 18;
    const int k = (c - e * 18) * 4;
#if USE_ASYNC_LDS
    __builtin_amdgcn_global_load_async_to_lds_b128(
        (AS1 v4i*)(gsrc + (size_t)c * 4), (AS3 v4i*)(&lds[e * ROWP + k]), 0, 0);
#else
    const float4 v = *(const float4*)(gsrc + (size_t)c * 4);
    *(float4*)(&lds[e * ROWP + k]) = v;
#endif
}

static __device__ __forceinline__ void stage_out(float* gdst, float* lds, int c) {
    const int e = c / 18;
    const int k = (c - e * 18) * 4;
#if USE_ASYNC_LDS
    __builtin_amdgcn_global_store_async_from_lds_b128(
        (AS1 v4i*)(gdst + (size_t)c * 4), (AS3 v4i*)(&lds[e * ROWP + k]), 0, 0);
#else
    *(float4*)(gdst + (size_t)c * 4) = *(const float4*)(&lds[e * ROWP + k]);
#endif
}

// FK chain for one element; reads pose from `row`, overwrites it in place with
// the posed joint positions (+trans). All tree indices compile-time constant.
static __device__ __forceinline__ void fk_element(float* row, const float* trans,
                                                  const float* Jm, int b) {
    const float tx = trans[(size_t)b * 3 + 0];
    const float ty = trans[(size_t)b * 3 + 1];
    const float tz = trans[(size_t)b * 3 + 2];

    float WR[kJ][9];   // world rotations (SROA'd: constant indices only)
    float Wt[kJ][3];   // world translations

#pragma unroll
    for (int j = 0; j < kJ; ++j) {
        // Rodrigues: axis-angle -> 3x3
        const float rx = row[3 * j + 0];
        const float ry = row[3 * j + 1];
        const float rz = row[3 * j + 2];
        const float t2  = rx * rx + ry * ry + rz * rz + 1e-16f;
        const float th  = sqrtf(t2);
        const float inv = 1.0f / th;
        const float ux = rx * inv, uy = ry * inv, uz = rz * inv;
        const float cth = __cosf(th), sth = __sinf(th);
        const float oc = 1.0f - cth;
        float R[9];
        R[0] = cth + oc * ux * ux;  R[1] = oc * ux * uy - sth * uz;  R[2] = oc * ux * uz + sth * uy;
        R[3] = oc * ux * uy + sth * uz;  R[4] = cth + oc * uy * uy;  R[5] = oc * uy * uz - sth * ux;
        R[6] = oc * ux * uz - sth * uy;  R[7] = oc * uy * uz + sth * ux;  R[8] = cth + oc * uz * uz;

        // local = J[j] @ [R 0; 0 1] : LR = J_R @ R, Lt = J_t
        // J is wave-uniform -> scalar (SMEM) loads.
        const float* Jr = Jm + j * 16;
        float LR[9], Lt[3];
#pragma unroll
        for (int r = 0; r < 3; ++r) {
            const float a0 = Jr[r * 4 + 0], a1 = Jr[r * 4 + 1], a2 = Jr[r * 4 + 2];
            LR[r * 3 + 0] = a0 * R[0] + a1 * R[3] + a2 * R[6];
            LR[r * 3 + 1] = a0 * R[1] + a1 * R[4] + a2 * R[7];
            LR[r * 3 + 2] = a0 * R[2] + a1 * R[5] + a2 * R[8];
            Lt[r] = Jr[r * 4 + 3];
        }

        if (j == 0) {
#pragma unroll
            for (int q = 0; q < 9; ++q) WR[0][q] = LR[q];
#pragma unroll
            for (int q = 0; q < 3; ++q) Wt[0][q] = Lt[q];
        } else {
            const int p = kPar[j];          // compile-time constant
#pragma unroll
            for (int r = 0; r < 3; ++r) {
                const float a0 = WR[p][r * 3 + 0];
                const float a1 = WR[p][r * 3 + 1];
                const float a2 = WR[p][r * 3 + 2];
                WR[j][r * 3 + 0] = a0 * LR[0] + a1 * LR[3] + a2 * LR[6];
                WR[j][r * 3 + 1] = a0 * LR[1] + a1 * LR[4] + a2 * LR[7];
                WR[j][r * 3 + 2] = a0 * LR[2] + a1 * LR[5] + a2 * LR[8];
                Wt[j][r] = a0 * Lt[0] + a1 * Lt[1] + a2 * Lt[2] + Wt[p][r];
            }
        }
        // Overwrite this row's pose slot j (read only at iteration j, only by
        // this thread) with the posed position.
        row[3 * j + 0] = Wt[j][0] + tx;
        row[3 * j + 1] = Wt[j][1] + ty;
        row[3 * j + 2] = Wt[j][2] + tz;
    }
}

// ---- hot path: every block is a full 128-element slab ----------------------
__global__ __launch_bounds__(EPB) void fk_kernel_full(const float* __restrict__ pose,
                                                      const float* __restrict__ trans,
                                                      const float* __restrict__ Jm,
                                                      float* __restrict__ out) {
    __shared__ float lds[EPB * ROWP];   // 128*76*4 = 38912 B

    const int tid  = threadIdx.x;
    const int base = blockIdx.x * EPB;

    const float* gsrc = pose + (size_t)base * ROWF;
#pragma unroll
    for (int i = 0; i < CPT; ++i) stage_in(gsrc, lds, tid + EPB * i);
#if USE_ASYNC_LDS
    wait_async0();
#endif
    __syncthreads();

    fk_element(&lds[tid * ROWP], trans, Jm, base + tid);
    __syncthreads();

    float* gdst = out + (size_t)base * ROWF;
#pragma unroll
    for (int i = 0; i < CPT; ++i) stage_out(gdst, lds, tid + EPB * i);
#if USE_ASYNC_LDS
    wait_async0();   // ensure async stores retire before wave end
#endif
}

// ---- remainder path: one guarded block (not launched when B % 128 == 0) ----
__global__ __launch_bounds__(EPB) void fk_kernel_tail(const float* __restrict__ pose,
                                                      const float* __restrict__ trans,
                                                      const float* __restrict__ Jm,
                                                      float* __restrict__ out,
                                                      int base, int nElem) {
    __shared__ float lds[EPB * ROWP];

    const int tid    = threadIdx.x;
    const int nChunk = nElem * (ROWF / 4);

    const float* gsrc = pose + (size_t)base * ROWF;
#pragma unroll
    for (int i = 0; i < CPT; ++i) {
        const int c = tid + EPB * i;
        if (c < nChunk) stage_in(gsrc, lds, c);
    }
#if USE_ASYNC_LDS
    wait_async0();
#endif
    __syncthreads();

    if (tid < nElem) fk_element(&lds[tid * ROWP], trans, Jm, base + tid);
    __syncthreads();

    float* gdst = out + (size_t)base * ROWF;
#pragma unroll
    for (int i = 0; i < CPT; ++i) {
        const int c = tid + EPB * i;
        if (c < nChunk) stage_out(gdst, lds, c);
    }
#if USE_ASYNC_LDS
    wait_async0();
#endif
}

extern "C" void kernel_launch(void* const* d_in, const int* in_sizes, int n_in,
                              void* d_out, int out_size, void* d_ws, size_t ws_size,
                              hipStream_t stream) {
    const float* pose  = (const float*)d_in[0];   // [B,24,3]
    const float* trans = (const float*)d_in[1];   // [B,3]
    const float* Jm    = (const float*)d_in[2];   // [24,4,4]
    // d_in[3] (parents) unused: tree is static, hardcoded (matches reference).
    float* out = (float*)d_out;                   // [B,24,3]

    const int batch = in_sizes[0] / (kJ * 3);
    const int nFull = batch / EPB;                // 1024 for B = 131072
    const int rem   = batch - nFull * EPB;        // 0 for B = 131072

    if (nFull > 0)
        fk_kernel_full<<<nFull, EPB, 0, stream>>>(pose, trans, Jm, out);
    if (rem > 0)
        fk_kernel_tail<<<1, EPB, 0, stream>>>(pose, trans, Jm, out,
                                              nFull * EPB, rem);
}